// MulModalityMaeMoe_78348793414342
// MI455X (gfx1250) — compile-verified
//
#include <hip/hip_runtime.h>
#include <math.h>

typedef float v2f __attribute__((ext_vector_type(2)));
typedef float v8f __attribute__((ext_vector_type(8)));

#define NL 10          // frames per modality (b*l inside space blocker / seq len)
#define DMODEL 3072
#define DINNER 6144
#define DSTATE 16
#define DTRANK 192
#define XDBN   224     // DTRANK + 2*DSTATE

// ---------------------------------------------------------------------------
// Generic fp32 WMMA GEMM:  C(M,N) = act( A(M,K; lda) @ W(N,K)^T + bias ) * mask
// One wave per 16x16 tile of C, K-loop in steps of 4 via V_WMMA_F32_16X16X4_F32.
// N must be a multiple of 16 and K a multiple of 4 (true for every layer here).
// M may be ragged (M=10 layers): loads clamp the row (EXEC stays all-ones for
// WMMA), stores are guarded.
// act: 0=none, 1=tanh, 2=softplus
// ---------------------------------------------------------------------------
__global__ __launch_bounds__(32)
void mm_gemm_wmma(const float* __restrict__ A, int lda,
                  const float* __restrict__ Wt,       // (N,K) row-major
                  const float* __restrict__ bias,     // nullable
                  float* __restrict__ C,
                  int M, int N, int K, int act,
                  const int* __restrict__ mask)       // nullable, scale = mask[0]
{
    const int lane = threadIdx.x;
    const int lo = lane & 15;
    const int hi = lane >> 4;
    const int n0 = blockIdx.x * 16;
    const int m0 = blockIdx.y * 16;

    int rowA = m0 + lo; if (rowA >= M) rowA = M - 1;   // clamp: keeps EXEC full
    int colB = n0 + lo; if (colB >= N) colB = N - 1;

    const float* Ap = A  + (size_t)rowA * lda + 2 * hi;
    const float* Bp = Wt + (size_t)colB * K   + 2 * hi;

    v8f acc = {};
    for (int k = 0; k < K; k += 4) {
        v2f a, b;
        a.x = Ap[k];  a.y = Ap[k + 1];                 // A[row][k+2*hi .. +1]
        b.x = Bp[k];  b.y = Bp[k + 1];                 // W[col][k+2*hi .. +1] = B^T
        acc = __builtin_amdgcn_wmma_f32_16x16x4_f32(
                  false, a, false, b, (short)0, acc, false, false);
    }

    const float scale = mask ? (float)mask[0] : 1.0f;
    const int cidx = n0 + lo;
    #pragma unroll
    for (int j = 0; j < 8; ++j) {
        int r = m0 + j + 8 * hi;                       // C/D layout: vgpr j -> M=j(+8)
        if (r < M && cidx < N) {
            float v = acc[j];
            if (bias) v += bias[cidx];
            if (act == 1)       v = tanhf(v);
            else if (act == 2)  v = (v > 20.f) ? v : log1pf(__expf(v));
            C[(size_t)r * N + cidx] = v * scale;
        }
    }
}

// ---- modality slice: x(3,40,128,128) -> img(10,3,128,128) ------------------
__global__ void mm_slice(const float* __restrict__ x, float* __restrict__ img, int base_l)
{
    long i = blockIdx.x * (long)blockDim.x + threadIdx.x;
    const long total = (long)NL * 3 * 16384;
    if (i >= total) return;
    int p = (int)(i % 16384); long t = i / 16384;
    int c = (int)(t % 3);     int l = (int)(t / 3);
    img[i] = x[((size_t)c * 40 + base_l + l) * 16384 + p];
}

// ---- (b,C,H,W) -> tokens (S*b, E), S=C*nh*nw, E=ph*pw ----------------------
__global__ void mm_im2tok(const float* __restrict__ img, float* __restrict__ tok,
                          int b, int C, int H, int W, int ph, int pw)
{
    int nh = H / ph, nw = W / pw, E = ph * pw;
    long total = (long)C * nh * nw * b * E;
    long i = blockIdx.x * (long)blockDim.x + threadIdx.x;
    if (i >= total) return;
    int e = (int)(i % E); long r = i / E;
    int bb = (int)(r % b); long s = r / b;
    int ix = (int)(s % nw); long t = s / nw;
    int iy = (int)(t % nh); int c = (int)(t / nh);
    int py = e / pw, px = e % pw;
    tok[i] = img[((size_t)(bb * C + c) * H + (iy * ph + py)) * W + ix * pw + px];
}

// ---- streaming-softmax attention: one thread per (bb, head, query) ---------
template <int HD>
__global__ void mm_attn(const float* __restrict__ qkv, float* __restrict__ o,
                        int S, int b, int E, int heads, float rs)
{
    long idx = blockIdx.x * (long)blockDim.x + threadIdx.x;
    long total = (long)b * heads * S;
    if (idx >= total) return;
    int s  = (int)(idx % S); long t = idx / S;
    int h  = (int)(t % heads);
    int bb = (int)(t / heads);
    const int E3 = 3 * E;
    float q[HD], acc[HD];
    #pragma unroll
    for (int d = 0; d < HD; ++d) {
        q[d] = qkv[(size_t)(s * b + bb) * E3 + h * HD + d] * rs;
        acc[d] = 0.f;
    }
    float mx = -1e30f, lsum = 0.f;
    for (int tt = 0; tt < S; ++tt) {
        const float* base = qkv + (size_t)(tt * b + bb) * E3 + h * HD;
        float sc = 0.f;
        #pragma unroll
        for (int d = 0; d < HD; ++d) sc += q[d] * base[E + d];
        float nm   = fmaxf(mx, sc);
        float corr = __expf(mx - nm);
        float pw   = __expf(sc - nm);
        lsum = lsum * corr + pw;
        #pragma unroll
        for (int d = 0; d < HD; ++d) acc[d] = acc[d] * corr + pw * base[2 * E + d];
        mx = nm;
    }
    float inv = 1.f / lsum;
    #pragma unroll
    for (int d = 0; d < HD; ++d)
        o[(size_t)(s * b + bb) * E + h * HD + d] = acc[d] * inv;
}

// ---- tokens back to image + residual ---------------------------------------
__global__ void mm_tok2img_res(const float* __restrict__ proj, const float* __restrict__ img,
                               float* __restrict__ y, int b, int C, int H, int W, int ph, int pw)
{
    long total = (long)b * C * H * W;
    long i = blockIdx.x * (long)blockDim.x + threadIdx.x;
    if (i >= total) return;
    int xx = (int)(i % W); long t = i / W;
    int yy = (int)(t % H); t /= H;
    int c  = (int)(t % C); int bb = (int)(t / C);
    int nh = H / ph, nw = W / pw, E = ph * pw;
    int iy = yy / ph, py = yy % ph, ix = xx / pw, px = xx % pw;
    long s = ((long)c * nh + iy) * nw + ix;
    y[i] = proj[(size_t)(s * b + bb) * E + py * pw + px] + img[i];
}

// ---- per-channel batch stats (mean over b,h,w) -----------------------------
__global__ void mm_chan_stats(const float* __restrict__ y, float* __restrict__ stats,
                              int b, int C, int HW)
{
    int c = blockIdx.x;
    __shared__ float ss[256], sq[256];
    float s = 0.f, q = 0.f;
    for (long i = threadIdx.x; i < (long)b * HW; i += blockDim.x) {
        int bb = (int)(i / HW), p = (int)(i % HW);
        float v = y[((size_t)bb * C + c) * HW + p];
        s += v; q += v * v;
    }
    ss[threadIdx.x] = s; sq[threadIdx.x] = q;
    __syncthreads();
    for (int st = 128; st > 0; st >>= 1) {
        if ((int)threadIdx.x < st) { ss[threadIdx.x] += ss[threadIdx.x + st];
                                     sq[threadIdx.x] += sq[threadIdx.x + st]; }
        __syncthreads();
    }
    if (threadIdx.x == 0) {
        float n = (float)b * HW;
        float mu = ss[0] / n;
        stats[2 * c] = mu;
        stats[2 * c + 1] = sq[0] / n - mu * mu;
    }
}

__global__ void mm_bn_tanh(float* __restrict__ y, const float* __restrict__ stats,
                           const float* __restrict__ g, const float* __restrict__ beta,
                           int b, int C, int HW)
{
    long total = (long)b * C * HW;
    long i = blockIdx.x * (long)blockDim.x + threadIdx.x;
    if (i >= total) return;
    int c = (int)((i / HW) % C);
    float mu = stats[2 * c], var = stats[2 * c + 1];
    y[i] = tanhf((y[i] - mu) * rsqrtf(var + 1e-5f) * g[c] + beta[c]);
}

// ---- direct conv2d (handles the concat([x,x]) duplication via ic % Cin_real)
__global__ void mm_conv2d(const float* __restrict__ in, const float* __restrict__ w,
                          const float* __restrict__ bias, float* __restrict__ out,
                          int Nimg, int Cin_eff, int Cin_real, int Hin, int Win,
                          int OC, int Hout, int Wout, int ksz, int stride, int pad)
{
    long total = (long)Nimg * OC * Hout * Wout;
    long i = blockIdx.x * (long)blockDim.x + threadIdx.x;
    if (i >= total) return;
    int ox = (int)(i % Wout); long t = i / Wout;
    int oy = (int)(t % Hout); t /= Hout;
    int oc = (int)(t % OC);   int n = (int)(t / OC);
    float acc = bias[oc];
    for (int ic = 0; ic < Cin_eff; ++ic) {
        int icr = ic % Cin_real;
        const float* ip = in + ((size_t)n * Cin_real + icr) * Hin * Win;
        const float* wp = w + ((size_t)oc * Cin_eff + ic) * ksz * ksz;
        for (int ky = 0; ky < ksz; ++ky) {
            int iy = oy * stride - pad + ky;
            if (iy < 0 || iy >= Hin) continue;
            for (int kx = 0; kx < ksz; ++kx) {
                int ix = ox * stride - pad + kx;
                if (ix < 0 || ix >= Win) continue;
                acc += ip[iy * Win + ix] * wp[ky * ksz + kx];
            }
        }
    }
    out[i] = acc;
}

// ---- mamba: causal depthwise conv (K=4) + SiLU -----------------------------
__global__ void mm_mamba_conv(const float* __restrict__ xz, const float* __restrict__ cw,
                              const float* __restrict__ cb, float* __restrict__ xs)
{
    long i = blockIdx.x * (long)blockDim.x + threadIdx.x;
    if (i >= (long)NL * DINNER) return;
    int d = (int)(i % DINNER), l = (int)(i / DINNER);
    float acc = cb[d];
    #pragma unroll
    for (int j = 0; j < 4; ++j) {
        int ll = l - 3 + j;
        if (ll >= 0) acc += xz[(size_t)ll * (2 * DINNER) + d] * cw[d * 4 + j];
    }
    xs[i] = acc / (1.f + __expf(-acc));
}

// ---- mamba selective scan (fused with D skip + z-gate) ---------------------
__global__ void mm_mamba_scan(const float* __restrict__ dt, const float* __restrict__ xdb,
                              const float* __restrict__ xs, const float* __restrict__ xz,
                              const float* __restrict__ A_log, const float* __restrict__ Dp,
                              float* __restrict__ y)
{
    int d = blockIdx.x * blockDim.x + threadIdx.x;
    if (d >= DINNER) return;
    float A[DSTATE], h[DSTATE];
    #pragma unroll
    for (int s = 0; s < DSTATE; ++s) { A[s] = -__expf(A_log[d * DSTATE + s]); h[s] = 0.f; }
    float Dd = Dp[d];
    for (int l = 0; l < NL; ++l) {
        float dtv = dt[(size_t)l * DINNER + d];
        float xv  = xs[(size_t)l * DINNER + d];
        const float* Bc = xdb + (size_t)l * XDBN + DTRANK;
        const float* Cc = Bc + DSTATE;
        float yv = 0.f;
        #pragma unroll
        for (int s = 0; s < DSTATE; ++s) {
            h[s] = __expf(dtv * A[s]) * h[s] + dtv * Bc[s] * xv;
            yv += h[s] * Cc[s];
        }
        yv += xv * Dd;
        float zv = xz[(size_t)l * (2 * DINNER) + DINNER + d];
        y[(size_t)l * DINNER + d] = yv * (zv / (1.f + __expf(-zv)));
    }
}

// ---- ConvTranspose (kd=1, per-depth-slice 2D), torch weight (IC,OC,1,kh,kw)
__global__ void mm_convt(const float* __restrict__ in, const float* __restrict__ w,
                         const float* __restrict__ bias, float* __restrict__ out,
                         int IC, int OC, int Dz, int Hin, int Win, int Hout, int Wout,
                         int ksz, int stride, int pad,
                         long out_c_stride, long out_z_stride, int do_tanh)
{
    long total = (long)OC * Dz * Hout * Wout;
    long i = blockIdx.x * (long)blockDim.x + threadIdx.x;
    if (i >= total) return;
    int ox = (int)(i % Wout); long t = i / Wout;
    int oy = (int)(t % Hout); t /= Hout;
    int z  = (int)(t % Dz);   int oc = (int)(t / Dz);
    float acc = bias[oc];
    for (int ic = 0; ic < IC; ++ic) {
        const float* ip = in + ((size_t)ic * Dz + z) * Hin * Win;
        const float* wp = w + ((size_t)ic * OC + oc) * ksz * ksz;
        for (int kh = 0; kh < ksz; ++kh) {
            int ty = oy + pad - kh;
            if (ty % stride != 0) continue;
            int iy = ty / stride;
            if (iy < 0 || iy >= Hin) continue;
            for (int kw = 0; kw < ksz; ++kw) {
                int tx = ox + pad - kw;
                if (tx % stride != 0) continue;
                int ix = tx / stride;
                if (ix < 0 || ix >= Win) continue;
                acc += ip[iy * Win + ix] * wp[kh * ksz + kw];
            }
        }
    }
    if (do_tanh) acc = tanhf(acc);
    out[(size_t)oc * out_c_stride + (size_t)z * out_z_stride + (size_t)oy * Wout + ox] = acc;
}

// ===========================================================================
static inline dim3 g1(long n) { return dim3((unsigned)((n + 255) / 256)); }

extern "C" void kernel_launch(void* const* d_in, const int* in_sizes, int n_in,
                              void* d_out, int out_size, void* d_ws, size_t ws_size,
                              hipStream_t stream)
{
    const float* x    = (const float*)d_in[0];
    const int*   mask = (const int*)d_in[1];
    int p = 2;
    // JAX pytree (sorted dict keys): dec, enc, ind, share
    const float* dec_p[4][10];   // d0_b,d0_w, d1_b,d1_w, d2_b,d2_w, t1_b,t1_w, t2_b,t2_w
    for (int i = 0; i < 4; ++i) for (int j = 0; j < 10; ++j) dec_p[i][j] = (const float*)d_in[p++];
    const float* enc_p[4][43];   // c1..c4(b,w), fc1(b,w), mamba(A_log,D,conv_b,conv_w,dt_b,dt_w,in_w,out_w,x_w), mha1..4(bn_b,bn_g,in_b,in_w,out_b,out_w)
    for (int i = 0; i < 4; ++i) for (int j = 0; j < 43; ++j) enc_p[i][j] = (const float*)d_in[p++];
    const float* ind_p[4][4];    // fc1_b, fc1_w, fc2_b, fc2_w
    for (int i = 0; i < 4; ++i) for (int j = 0; j < 4;  ++j) ind_p[i][j] = (const float*)d_in[p++];
    const float* sh_p[6];        // fc1_b, fc1_w, fc2_b, fc2_w, fc3_b, fc3_w
    for (int j = 0; j < 6; ++j) sh_p[j] = (const float*)d_in[p++];

    // ---- workspace arena (floats), ~20MB, reused across modalities --------
    float* Wk = (float*)d_ws;
    const size_t IMG = 491520;                       // (10,3,128,128)
    float* img0  = Wk;
    float* img1  = Wk + IMG;
    float* tok   = Wk + 2 * IMG;                     // tokens / proj (<=491520)
    float* qkvb  = Wk + 3 * IMG;                     // <= 7680*192
    float* ob    = qkvb + 1474560;                   // attn out (<=491520)
    float* stats = ob + IMG;                         // 128
    float* xz    = stats + 128;                      // 10*12288
    float* xs    = xz + 122880;                      // 10*6144
    float* xdb   = xs + 61440;                       // 10*224
    float* dtb   = xdb + 2240;                       // 10*6144
    float* ym    = dtb + 61440;                      // 10*6144
    float* mo    = ym + 61440;                       // 10*3072
    float* encb  = mo + 30720;                       // 10*1024
    float* s1    = encb + 10240;                     // 10*2048
    float* s2    = s1 + 20480;                       // 10*1024
    float* i1    = s2 + 10240;                       // 10*2048
    float* decin = i1 + 20480;                       // 20*1024
    float* tA    = decin + 20480;                    // 20*32*32
    float* tB    = tA + 20480;                       // 10*32*32
    float* d0b   = tB + 10240;                       // 2*10*64*64
    float* d1b   = d0b + 81920;                      // 6*10*128*128

    const int phs[4]   = {8, 8, 4, 4};
    const int heads[4] = {8, 8, 16, 16};

    for (int m = 0; m < 4; ++m) {
        // ---- slice this modality's 10 frames ------------------------------
        mm_slice<<<g1((long)NL * 3 * 16384), 256, 0, stream>>>(x, img0, m * 10);

        // ---- space blocker: 4x (MHA block + conv k4 s2 p1) ----------------
        int C = 3, H = 128;
        for (int s = 0; s < 4; ++s) {
            const int ph = phs[s], hds = heads[s];
            const int E = ph * ph, nh = H / ph;
            const int S = C * nh * nh, R = S * NL, hd = E / hds;
            const float* bn_b  = enc_p[m][19 + 6 * s + 0];
            const float* bn_g  = enc_p[m][19 + 6 * s + 1];
            const float* in_b  = enc_p[m][19 + 6 * s + 2];
            const float* in_w  = enc_p[m][19 + 6 * s + 3];
            const float* out_b = enc_p[m][19 + 6 * s + 4];
            const float* out_w = enc_p[m][19 + 6 * s + 5];

            mm_im2tok<<<g1((long)R * E), 256, 0, stream>>>(img0, tok, NL, C, H, H, ph, ph);
            mm_gemm_wmma<<<dim3((3 * E) / 16, (R + 15) / 16), 32, 0, stream>>>(
                tok, E, in_w, in_b, qkvb, R, 3 * E, E, 0, nullptr);
            if (hd == 8)
                mm_attn<8><<<g1((long)NL * hds * S), 256, 0, stream>>>(
                    qkvb, ob, S, NL, E, hds, rsqrtf((float)hd));
            else
                mm_attn<1><<<g1((long)NL * hds * S), 256, 0, stream>>>(
                    qkvb, ob, S, NL, E, hds, rsqrtf((float)hd));
            mm_gemm_wmma<<<dim3(E / 16, (R + 15) / 16), 32, 0, stream>>>(
                ob, E, out_w, out_b, tok, R, E, E, 0, nullptr);
            mm_tok2img_res<<<g1((long)NL * C * H * H), 256, 0, stream>>>(
                tok, img0, img1, NL, C, H, H, ph, ph);
            mm_chan_stats<<<C, 256, 0, stream>>>(img1, stats, NL, C, H * H);
            mm_bn_tanh<<<g1((long)NL * C * H * H), 256, 0, stream>>>(
                img1, stats, bn_g, bn_b, NL, C, H * H);

            const int OC = 2 * C, Hout = H / 2;
            mm_conv2d<<<g1((long)NL * OC * Hout * Hout), 256, 0, stream>>>(
                img1, enc_p[m][2 * s + 1], enc_p[m][2 * s], img0,
                NL, 2 * C, C, H, H, OC, Hout, Hout, 4, 2, 1);
            C = OC; H = Hout;
        }
        // img0 now holds (10,48,8,8) == contiguous (10, 3072) sequence

        // ---- mamba --------------------------------------------------------
        mm_gemm_wmma<<<dim3(12288 / 16, 1), 32, 0, stream>>>(
            img0, DMODEL, enc_p[m][16], nullptr, xz, NL, 2 * DINNER, DMODEL, 0, nullptr);
        mm_mamba_conv<<<g1((long)NL * DINNER), 256, 0, stream>>>(
            xz, enc_p[m][13], enc_p[m][12], xs);
        mm_gemm_wmma<<<dim3(XDBN / 16, 1), 32, 0, stream>>>(
            xs, DINNER, enc_p[m][18], nullptr, xdb, NL, XDBN, DINNER, 0, nullptr);
        mm_gemm_wmma<<<dim3(DINNER / 16, 1), 32, 0, stream>>>(
            xdb, XDBN, enc_p[m][15], enc_p[m][14], dtb, NL, DINNER, DTRANK, 2, nullptr);
        mm_mamba_scan<<<g1(DINNER), 256, 0, stream>>>(
            dtb, xdb, xs, xz, enc_p[m][10], enc_p[m][11], ym);
        mm_gemm_wmma<<<dim3(DMODEL / 16, 1), 32, 0, stream>>>(
            ym, DINNER, enc_p[m][17], nullptr, mo, NL, DMODEL, DINNER, 0, nullptr);
        // fc1 + tanh + mask
        mm_gemm_wmma<<<dim3(1024 / 16, 1), 32, 0, stream>>>(
            mo, DMODEL, enc_p[m][9], enc_p[m][8], encb, NL, 1024, DMODEL, 1, mask + m);

        // ---- share / indep heads (write straight into decoder input) ------
        mm_gemm_wmma<<<dim3(2048 / 16, 1), 32, 0, stream>>>(
            encb, 1024, sh_p[1], sh_p[0], s1, NL, 2048, 1024, 0, nullptr);
        mm_gemm_wmma<<<dim3(1024 / 16, 1), 32, 0, stream>>>(
            s1, 2048, sh_p[3], sh_p[2], s2, NL, 1024, 2048, 0, nullptr);
        mm_gemm_wmma<<<dim3(1024 / 16, 1), 32, 0, stream>>>(
            s2, 1024, sh_p[5], sh_p[4], decin, NL, 1024, 1024, 1, mask + m);
        mm_gemm_wmma<<<dim3(2048 / 16, 1), 32, 0, stream>>>(
            encb, 1024, ind_p[m][1], ind_p[m][0], i1, NL, 2048, 1024, 0, nullptr);
        mm_gemm_wmma<<<dim3(1024 / 16, 1), 32, 0, stream>>>(
            i1, 2048, ind_p[m][3], ind_p[m][2], decin + 10240, NL, 1024, 2048, 1, mask + m);

        // ---- decoder ------------------------------------------------------
        mm_conv2d<<<g1((long)20 * 32 * 32), 256, 0, stream>>>(
            decin, dec_p[m][7], dec_p[m][6], tA, 1, 20, 20, 32, 32, 20, 32, 32, 3, 1, 1);
        mm_conv2d<<<g1((long)10 * 32 * 32), 256, 0, stream>>>(
            tA, dec_p[m][9], dec_p[m][8], tB, 1, 20, 20, 32, 32, 10, 32, 32, 3, 1, 1);
        mm_convt<<<g1((long)2 * 10 * 64 * 64), 256, 0, stream>>>(
            tB, dec_p[m][1], dec_p[m][0], d0b, 1, 2, 10, 32, 32, 64, 64, 4, 2, 1,
            (long)10 * 64 * 64, (long)64 * 64, 0);
        mm_convt<<<g1((long)6 * 10 * 128 * 128), 256, 0, stream>>>(
            d0b, dec_p[m][3], dec_p[m][2], d1b, 2, 6, 10, 64, 64, 128, 128, 4, 2, 1,
            (long)10 * 128 * 128, (long)128 * 128, 0);
        // final convT + tanh written directly into d_out at depth offset m*10
        mm_convt<<<g1((long)3 * 10 * 128 * 128), 256, 0, stream>>>(
            d1b, dec_p[m][5], dec_p[m][4],
            (float*)d_out + (size_t)m * 10 * 16384,
            6, 3, 10, 128, 128, 128, 128, 3, 1, 1,
            (long)40 * 16384, (long)16384, 1);
    }
    (void)in_sizes; (void)n_in; (void)out_size; (void)ws_size;
}